// GMP_70334384439822
// MI455X (gfx1250) — compile-verified
//
#include <hip/hip_runtime.h>
#include <hip/hip_bf16.h>

// GMP (generalized memory polynomial) as a skinny fp32 GEMM on CDNA5 matrix
// cores:  coef(16x16) = W(16x80) @ F(80x16 samples), 20x V_WMMA_F32_16X16X4_F32.
// Features per sample: [1, p0..p0^3, p1..p1^4, ..., p19..p19^4], p[d]=|x[s-d]|.
//
// v2: the feature pipeline is staged cooperatively.  Each wave owns a strip of
// 128 samples; once per strip the 32 lanes build the 147-value magnitude/power
// window {x, p, p^2, p^3, p^4} in LDS (raw v_sqrt_f32, ~5 values per lane).
// The per-tile hot loop is then pure data movement + matrix ops: each B
// fragment is one aligned ds_load_b64 at a compile-time DS offset, and 20
// chained WMMAs produce all 10 coefficients for 16 samples.

typedef __attribute__((ext_vector_type(2))) float v2f;
typedef __attribute__((ext_vector_type(8))) float v8f;

#define Kp 4
#define Lp 10
#define Mp 10
#define Dp 20            // L+M memory depth
#define NF 80            // features = 20 chunks of K=4
#define NCHUNK 20
#define S_ 16384         // power of two: batch = flat >> 14, s = flat & (S_-1)
#define S_SHIFT 14
#define WAVES_PER_BLOCK 8
#define TILES_PER_WAVE 8
#define STRIP 128                       // samples per wave strip
#define NWIN 160                        // >= STRIP + Dp - 1 = 147, 32-multiple

__global__ __launch_bounds__(256) void gmp_wmma_kernel(
    const float* __restrict__ x,   // (B, S, 2)
    const float* __restrict__ a,   // (K, L)
    const float* __restrict__ b,   // (K, L, M)
    float* __restrict__ out,       // (B, S, 2)
    int ntiles) {
  __shared__ float W[16 * NF];                                    // 5 KB
  __shared__ __align__(8)  float xw[WAVES_PER_BLOCK][NWIN * 2];   // 10 KB
  __shared__ __align__(16) float pt[WAVES_PER_BLOCK][NWIN * 4];   // 20 KB

  const int tid  = threadIdx.x;
  const int lane = tid & 31;
  const int wave = tid >> 5;
  const int col  = lane & 15;      // sample column within 16-wide tile
  const int half = lane >> 4;      // K-pair selector (ISA 16x4 / 4x16 layout)

  // ---- Build W (16 x 80) in LDS, once per block ----
  for (int i = tid; i < 16 * NF; i += 256) W[i] = 0.0f;
  __syncthreads();
  if (tid < Lp) {
    const int l = tid;                        // one thread per row -> no races
    float* row = &W[l * NF];
    row[0] += a[0 * Lp + (9 - l)];            // coef1 k=0 (0^0==1 matches ref)
    for (int k = 1; k <= 3; ++k) {
      const int f = (l == 0) ? k : (4 * l + k - 1);
      row[f] += a[k * Lp + (9 - l)];          // coef1: a[k, 9-l] on p[l]^k
    }
    for (int k = 1; k <= Kp; ++k)             // coef2: b[k-1,l,m] on p[l+1+m]^k
      for (int m = 0; m < Mp; ++m) {
        const int d = l + 1 + m;              // 1..19
        row[4 * d + k - 1] += b[(k - 1) * Lp * Mp + l * Mp + m];
      }
  }

  // ---- Stage per-wave magnitude/power window for this wave's 128-sample strip
  const int tile0 = (blockIdx.x * WAVES_PER_BLOCK + wave) * TILES_PER_WAVE;
  const bool strip_ok = tile0 < ntiles;
  const unsigned flat0 = (unsigned)tile0 * 16u;   // strip start, 128-aligned
  const int batch = (int)(flat0 >> S_SHIFT);
  const int s0    = (int)(flat0 & (S_ - 1));      // strip start within batch
  const float* xb = x + (size_t)batch * (size_t)S_ * 2;

  for (int j = 0; j < NWIN / 32; ++j) {
    const int i  = lane + 32 * j;                 // window slot
    const int sv = s0 - (Dp - 1) + i;             // sample index within batch
    float re = 0.0f, im = 0.0f;
    if (strip_ok && sv >= 0 && sv < S_) {
      re = xb[2 * sv];
      im = xb[2 * sv + 1];
    }
    const float p  = __builtin_amdgcn_sqrtf(re * re + im * im); // raw v_sqrt_f32
    const float p2 = p * p;
    float2* xq = (float2*)&xw[wave][2 * i];
    *xq = make_float2(re, im);                    // ds_store_b64
    float4* pq = (float4*)&pt[wave][4 * i];
    *pq = make_float4(p, p2, p2 * p, p2 * p2);    // ds_store_b128
  }
  __syncthreads();

  // ---- Preload loop-invariant A fragments: lane holds W[col, 4c+2h .. +1] ----
  v2f afrag[NCHUNK];
  {
    const float* row = &W[col * NF];
    for (int c = 0; c < NCHUNK; ++c) {
      afrag[c].x = row[4 * c + 2 * half];
      afrag[c].y = row[4 * c + 2 * half + 1];
    }
  }

  for (int t = 0; t < TILES_PER_WAVE; ++t) {
    const int tile = tile0 + t;
    if (tile >= ntiles) break;           // uniform across wave: EXEC stays all-1s
    const int ib = (Dp - 1) + 16 * t + col;   // window slot of this sample
    const int s  = s0 + 16 * t + col;         // sample within batch

    // ---- coef = W @ features : 20 chained f32 WMMAs, B frags from LDS ----
    v8f acc = {};
    {   // chunk 0: features {1, p0, p0^2, p0^3}
      const float vA = pt[wave][4 * ib + half];       // h0: p,  h1: p^2
      const float vB = pt[wave][4 * ib + half + 1];   // h0: -,  h1: p^3
      v2f bf;
      bf.x = half ? vA : 1.0f;
      bf.y = half ? vB : vA;
      acc = __builtin_amdgcn_wmma_f32_16x16x4_f32(false, afrag[0], false, bf,
                                                  (short)0, acc, false, false);
    }
    for (int c = 1; c < NCHUNK; ++c) {  // chunk c: features {p[c]^1..^4}
      const float2 bq =
          *(const float2*)&pt[wave][4 * (ib - c) + 2 * half];  // ds_load_b64
      v2f bf;
      bf.x = bq.x;
      bf.y = bq.y;
      acc = __builtin_amdgcn_wmma_f32_16x16x4_f32(false, afrag[c], false, bf,
                                                  (short)0, acc, false, false);
    }

    // ---- y[s] = sum_l coef[l,s] * x[s-l]; rows split across lane halves ----
    float yr = 0.0f, yi = 0.0f;
    const int dbase = 8 * half;          // lanes 0-15: l=0..7, 16-31: l=8..15
    for (int jj = 0; jj < 8; ++jj) {
      const float cv = acc[jj];          // rows 10..15 of W are zero
      const float2 xv = *(const float2*)&xw[wave][2 * (ib - dbase - jj)];
      yr += cv * xv.x;
      yi += cv * xv.y;
    }
    yr += __shfl_down(yr, 16, 32);       // combine the two half-sums per sample
    yi += __shfl_down(yi, 16, 32);
    if (half == 0) {
      float r0 = yr, i0 = yi;
      if (s < Dp) { r0 = 0.0f; i0 = 0.0f; }   // reference mask: s >= D
      float* o = out + (((size_t)batch << S_SHIFT) + (size_t)s) * 2;
      o[0] = r0;
      o[1] = i0;
    }
  }
}

extern "C" void kernel_launch(void* const* d_in, const int* in_sizes, int n_in,
                              void* d_out, int out_size, void* d_ws, size_t ws_size,
                              hipStream_t stream) {
  const float* x = (const float*)d_in[0];   // (32, 16384, 2) fp32
  const float* a = (const float*)d_in[1];   // (4, 10) fp32
  const float* b = (const float*)d_in[2];   // (4, 10, 10) fp32   (d_in[3]=c unused)
  float* out = (float*)d_out;               // (32, 16384, 2) fp32

  const int nsamples = out_size / 2;                                    // 524288
  const int ntiles   = (nsamples + 15) / 16;                            // 32768
  const int waves    = (ntiles + TILES_PER_WAVE - 1) / TILES_PER_WAVE;  // 4096
  const int blocks   = (waves + WAVES_PER_BLOCK - 1) / WAVES_PER_BLOCK; // 512

  gmp_wmma_kernel<<<blocks, 256, 0, stream>>>(x, a, b, out, ntiles);
}